// OnlineTripletLossV3_29171417874648
// MI455X (gfx1250) — compile-verified
//
#include <hip/hip_runtime.h>
#include <stdint.h>

// ---------------------------------------------------------------------------
// Online triplet loss on MI455X (gfx1250, wave32).
//
// Strategy: for each group of 16 triplets, build the 16xK matrix of
// difference vectors and compute Delta * Delta^T with the full-precision
// V_WMMA_F32_16X16X4_F32 instruction, accumulating over K=256 in chunks of 4.
// The diagonal of the 16x16 f32 accumulator gives the 16 squared L2
// distances. Because the product is symmetric and the documented A(16x4) and
// B(4x16) f32 lane layouts mirror each other, the SAME v2f register pair
// serves as both the A and B operand -- one float2 load per matrix per chunk.
// ---------------------------------------------------------------------------

typedef __attribute__((ext_vector_type(2))) float v2f;
typedef __attribute__((ext_vector_type(8))) float v8f;

#define TL_MARGIN 1.0f
#define TL_D 256

// Constant-index 8-way select (avoids dynamic vector indexing -> scratch).
__device__ __forceinline__ float vget8(v8f v, int r) {
  float x = v[0];
  x = (r == 1) ? v[1] : x;
  x = (r == 2) ? v[2] : x;
  x = (r == 3) ? v[3] : x;
  x = (r == 4) ? v[4] : x;
  x = (r == 5) ? v[5] : x;
  x = (r == 6) ? v[6] : x;
  x = (r == 7) ? v[7] : x;
  return x;
}

__global__ void triplet_wmma_partials(const float* __restrict__ emb,
                                      const long long* __restrict__ trip,
                                      float* __restrict__ partials,
                                      int T, int groups) {
  const int lane = threadIdx.x & 31;
  const int wave = (int)((blockIdx.x * blockDim.x + threadIdx.x) >> 5);
  if (wave >= groups) return;  // wave-uniform: EXEC stays all-ones for WMMA

  // Matrix row/col this lane contributes to (both halves of the wave cover
  // rows 0..15; lanes >=16 supply the K+2,K+3 slice per the ISA A-layout).
  const int j = lane & 15;
  long long t = (long long)wave * 16 + j;
  const bool tvalid = (t < (long long)T);
  if (!tvalid) t = (long long)T - 1;  // clamp via select; no EXEC divergence

  const long long* tp = trip + 3 * t;
  const long long ia = tp[0];
  const long long ip = tp[1];
  const long long in_ = tp[2];
  const float* __restrict__ pa = emb + ia * TL_D;
  const float* __restrict__ pp = emb + ip * TL_D;
  const float* __restrict__ pn = emb + in_ * TL_D;

  const int khalf = (lane >> 4) << 1;  // 0 for lanes 0-15, 2 for lanes 16-31

  v8f c_ap = {};
  v8f c_an = {};
  v8f c_pn = {};

#pragma unroll 4
  for (int k0 = 0; k0 < TL_D; k0 += 4) {
    const int k = k0 + khalf;
    // Rows are 1KB apart from a 256B-aligned base; k is even -> 8B aligned.
    const v2f a = *(const v2f*)(pa + k);
    const v2f p = *(const v2f*)(pp + k);
    const v2f n = *(const v2f*)(pn + k);
    const v2f dap = a - p;
    const v2f dan = a - n;
    const v2f dpn = p - n;
    // Symmetric outer-product accumulate: A and B operands are identical.
    c_ap = __builtin_amdgcn_wmma_f32_16x16x4_f32(false, dap, false, dap,
                                                 (short)0, c_ap, false, false);
    c_an = __builtin_amdgcn_wmma_f32_16x16x4_f32(false, dan, false, dan,
                                                 (short)0, c_an, false, false);
    c_pn = __builtin_amdgcn_wmma_f32_16x16x4_f32(false, dpn, false, dpn,
                                                 (short)0, c_pn, false, false);
  }

  // Diagonal element j of the 16x16 f32 C/D layout:
  //   j in [0,8):  lane j,      VGPR j
  //   j in [8,16): lane j + 16, VGPR j - 8
  // => owner lanes are {0..7} and {24..31}, component index = lane & 7.
  const bool owner = (lane < 8) || (lane >= 24);
  const int r = lane & 7;
  const float ap2 = vget8(c_ap, r);
  const float an2 = vget8(c_an, r);
  const float pn2 = vget8(c_pn, r);
  const float d_ap = __builtin_sqrtf(fmaxf(ap2, 0.0f));
  const float d_an = __builtin_sqrtf(fmaxf(an2, 0.0f));
  const float d_pn = __builtin_sqrtf(fmaxf(pn2, 0.0f));
  float loss = fmaxf(d_ap - fminf(d_an, d_pn) + TL_MARGIN, 0.0f);
  loss = (owner && tvalid) ? loss : 0.0f;

  // Wave32 butterfly reduction of the 16 owned losses.
#pragma unroll
  for (int m = 16; m >= 1; m >>= 1) loss += __shfl_xor(loss, m, 32);

  if (lane == 0) partials[wave] = loss;
}

// Deterministic fixed-order final reduction (single block, no float atomics,
// so graph replays are bitwise reproducible).
__global__ void triplet_finalize(const float* __restrict__ partials,
                                 float* __restrict__ out, int groups, int T) {
  __shared__ float sm[256];
  float s = 0.0f;
  for (int i = threadIdx.x; i < groups; i += 256) s += partials[i];
  sm[threadIdx.x] = s;
  __syncthreads();
#pragma unroll
  for (int w = 128; w >= 1; w >>= 1) {
    if ((int)threadIdx.x < w) sm[threadIdx.x] += sm[threadIdx.x + w];
    __syncthreads();
  }
  if (threadIdx.x == 0) {
    out[0] = sm[0] / (float)T;  // losses.mean()
    out[1] = (float)T;          // triplets.shape[0]
  }
}

extern "C" void kernel_launch(void* const* d_in, const int* in_sizes, int n_in,
                              void* d_out, int out_size, void* d_ws,
                              size_t ws_size, hipStream_t stream) {
  const float* emb = (const float*)d_in[0];            // (B, 256) f32
  const long long* trip = (const long long*)d_in[1];   // (T, 3) int64
  float* out = (float*)d_out;                          // [mean, T]
  float* partials = (float*)d_ws;                      // one f32 per group

  const int T = in_sizes[1] / 3;
  const int groups = (T + 15) / 16;  // 6250 for T=100000 (exact)

  const int wavesPerBlock = 8;  // 256 threads = 8 wave32 waves
  const int blocks = (groups + wavesPerBlock - 1) / wavesPerBlock;

  triplet_wmma_partials<<<blocks, 32 * wavesPerBlock, 0, stream>>>(
      emb, trip, partials, T, groups);
  triplet_finalize<<<1, 256, 0, stream>>>(partials, out, groups, T);
}